// InteractPre_43645457662720
// MI455X (gfx1250) — compile-verified
//
#include <hip/hip_runtime.h>
#include <hip/hip_bf16.h>
#include <math.h>

// ---------------------------------------------------------------------------
// Sizes (fixed by the reference)
// ---------------------------------------------------------------------------
#define QN 512
#define LN 4096

typedef __attribute__((ext_vector_type(2))) float v2f;
typedef __attribute__((ext_vector_type(8))) float v8f;

enum { ACT_NONE = 0, ACT_RELU = 1, ACT_SIGMOID = 2, ACT_LRELU = 3 };

// Register-blocked WMMA GEMM:  C[M,N] = act(A[M,K] @ W[N,K]^T + bias[N])
//
// One wave32 computes a 32x64 strip: TM=2 m-subtiles x TN=4 n-subtiles,
// 8 independent f32 accumulators. Per K-step (k+=4): 2 A-frag + 4 B-frag
// b64 loads feed 8 V_WMMA_F32_16X16X4_F32 ops -> 0.75 loads/WMMA (vs 2.0
// for an unblocked tile), and 8 independent acc chains for matrix-pipe ILP.
//
// f32 16x16x4 VGPR layouts (ISA 7.12.2):
//   A (16x4):  lanes 0-15 -> M=lane,    VGPR0=K0, VGPR1=K1
//              lanes 16-31-> M=lane-16, VGPR0=K2, VGPR1=K3
//   B (4x16):  lane n holds col N=n; VGPR0/1 = K{0,1} (lanes 0-15) or
//              K{2,3} (lanes 16-31).  B[k][n] == W[n][k] -> contiguous f2 load.
//   C/D:       VGPR v = row M=v (+8 for upper half-lanes), N = lane&15.
#define TM 2
#define TN 4

__global__ void wmma_gemm_bias_act(const float* __restrict__ A,
                                   const float* __restrict__ W,
                                   const float* __restrict__ bias,
                                   float* __restrict__ C,
                                   int M, int N, int K, int act)
{
    const int n0   = blockIdx.x * (16 * TN);
    const int m0   = blockIdx.y * (16 * TM);
    const int lane = threadIdx.x;      // 0..31 (wave32)
    const int half = lane >> 4;        // 0 or 1
    const int lid  = lane & 15;

    const float* __restrict__ arow = A + (size_t)(m0 + lid) * K + half * 2;
    const float* __restrict__ wrow = W + (size_t)(n0 + lid) * K + half * 2;

    v8f acc[TM][TN];
#pragma unroll
    for (int i = 0; i < TM; ++i)
#pragma unroll
        for (int j = 0; j < TN; ++j) acc[i][j] = (v8f){};

    for (int k0 = 0; k0 < K; k0 += 4) {
        v2f af[TM], bf[TN];
#pragma unroll
        for (int i = 0; i < TM; ++i)
            af[i] = *(const v2f*)(arow + (size_t)(16 * i) * K + k0);
#pragma unroll
        for (int j = 0; j < TN; ++j)
            bf[j] = *(const v2f*)(wrow + (size_t)(16 * j) * K + k0);
#pragma unroll
        for (int i = 0; i < TM; ++i)
#pragma unroll
            for (int j = 0; j < TN; ++j)
                // (neg_a, A, neg_b, B, c_mod, C, reuse_a, reuse_b)
                acc[i][j] = __builtin_amdgcn_wmma_f32_16x16x4_f32(
                    false, af[i], false, bf[j], (short)0, acc[i][j], false, false);
    }

#pragma unroll
    for (int j = 0; j < TN; ++j) {
        const int   n = n0 + 16 * j + lid;
        const float b = bias[n];
#pragma unroll
        for (int i = 0; i < TM; ++i) {
            const int row0 = m0 + 16 * i + half * 8;
#pragma unroll
            for (int v = 0; v < 8; ++v) {
                float x = acc[i][j][v] + b;
                if      (act == ACT_RELU)    x = fmaxf(x, 0.f);
                else if (act == ACT_SIGMOID) x = 1.f / (1.f + __expf(-x));
                else if (act == ACT_LRELU)   x = (x > 0.f) ? x : 0.01f * x;
                C[(size_t)(row0 + v) * N + n] = x;
            }
        }
    }
}

// ---------------------------------------------------------------------------
// out[o][c] = (1/n_in) * sum_i relu(inner[i][c] + outer[o][c]),  c in [0,64)
// 8 outer rows per block so each streamed inner element feeds 8 accumulators.
// ---------------------------------------------------------------------------
__global__ void relu_outer_mean(const float* __restrict__ inner, int n_in,
                                const float* __restrict__ outer,
                                float* __restrict__ out)
{
    const int o0    = blockIdx.x * 8;
    const int c     = threadIdx.x & 63;
    const int chunk = threadIdx.x >> 6;

    float ov[8], acc[8];
#pragma unroll
    for (int j = 0; j < 8; ++j) { ov[j] = outer[(size_t)(o0 + j) * 64 + c]; acc[j] = 0.f; }

    for (int i = chunk; i < n_in; i += 4) {
        const float iv = inner[(size_t)i * 64 + c];
#pragma unroll
        for (int j = 0; j < 8; ++j) acc[j] += fmaxf(iv + ov[j], 0.f);
    }

    __shared__ float red[4][64];
    const float inv = 1.0f / (float)n_in;
#pragma unroll
    for (int j = 0; j < 8; ++j) {
        red[chunk][c] = acc[j];
        __syncthreads();
        if (chunk == 0)
            out[(size_t)(o0 + j) * 64 + c] =
                (red[0][c] + red[1][c] + red[2][c] + red[3][c]) * inv;
        __syncthreads();
    }
}

// prot[c] = max_l ( k[l][c] * (1 + p_gate[l][c]) ), single block of 256.
__global__ void prot_max_kernel(const float* __restrict__ k,
                                const float* __restrict__ p_gate,
                                float* __restrict__ prot, int Lcnt)
{
    const int c     = threadIdx.x & 63;
    const int chunk = threadIdx.x >> 6;
    float m = -INFINITY;
    for (int l = chunk; l < Lcnt; l += 4) {
        const float kv = k[(size_t)l * 64 + c];
        m = fmaxf(m, kv + kv * p_gate[(size_t)l * 64 + c]);
    }
    __shared__ float red[4][64];
    red[chunk][c] = m;
    __syncthreads();
    if (chunk == 0)
        prot[c] = fmaxf(fmaxf(red[0][c], red[1][c]), fmaxf(red[2][c], red[3][c]));
}

// ctx[q][0:64] = r[q]*(1+r_gate[q]);  ctx[q][64:128] = prot
__global__ void ctx_kernel(const float* __restrict__ r,
                           const float* __restrict__ r_gate,
                           const float* __restrict__ prot,
                           float* __restrict__ ctx, int Q)
{
    const int idx = blockIdx.x * blockDim.x + threadIdx.x;
    const int q = idx >> 7, c = idx & 127;
    if (q >= Q) return;
    float v;
    if (c < 64) {
        const float rv = r[(size_t)q * 64 + c];
        v = rv + rv * r_gate[(size_t)q * 64 + c];
    } else {
        v = prot[c - 64];
    }
    ctx[(size_t)q * 128 + c] = v;
}

// out[q] = h2[q] . Wf3 + bf3   (128-wide dot per row)
__global__ void out_kernel(const float* __restrict__ h2,
                           const float* __restrict__ Wf3,
                           const float* __restrict__ bf3,
                           float* __restrict__ out, int Q)
{
    const int q = blockIdx.x * blockDim.x + threadIdx.x;
    if (q >= Q) return;
    float s = bf3[0];
#pragma unroll 4
    for (int c = 0; c < 128; ++c) s += h2[(size_t)q * 128 + c] * Wf3[c];
    out[q] = s;
}

// ---------------------------------------------------------------------------
// Launcher
// ---------------------------------------------------------------------------
static inline void gemm(const float* A, const float* W, const float* b, float* C,
                        int M, int N, int K, int act, hipStream_t s)
{
    dim3 grid(N / (16 * TN), M / (16 * TM));   // N % 64 == 0, M % 32 == 0 here
    wmma_gemm_bias_act<<<grid, 32, 0, s>>>(A, W, b, C, M, N, K, act);
}

extern "C" void kernel_launch(void* const* d_in, const int* in_sizes, int n_in,
                              void* d_out, int out_size, void* d_ws, size_t ws_size,
                              hipStream_t stream)
{
    const float* reactions = (const float*)d_in[0];  // [1,512,256]
    const float* protein   = (const float*)d_in[1];  // [1,4096,1024]
    const float* Wc  = (const float*)d_in[2];  const float* bc  = (const float*)d_in[3];
    const float* W1  = (const float*)d_in[4];  const float* b1  = (const float*)d_in[5];
    const float* W2  = (const float*)d_in[6];  const float* b2  = (const float*)d_in[7];
    const float* Wa  = (const float*)d_in[8];  const float* ba  = (const float*)d_in[9];
    const float* Wpa = (const float*)d_in[10]; const float* bpa = (const float*)d_in[11];
    const float* Wra = (const float*)d_in[12]; const float* bra = (const float*)d_in[13];
    const float* Wf1 = (const float*)d_in[14]; const float* bf1 = (const float*)d_in[15];
    const float* Wf2 = (const float*)d_in[16]; const float* bf2 = (const float*)d_in[17];
    const float* Wf3 = (const float*)d_in[18]; const float* bf3 = (const float*)d_in[19];

    // workspace layout (floats)
    float* ws = (float*)d_ws;
    float* p      = ws;                 ws += (size_t)LN * 256;  // 4 MB
    float* kbuf   = ws;                 ws += (size_t)LN * 64;   // 1 MB
    float* pa     = ws;                 ws += (size_t)LN * 64;   // 1 MB
    float* r      = ws;                 ws += (size_t)QN * 64;
    float* ra     = ws;                 ws += (size_t)QN * 64;
    float* S_r    = ws;                 ws += (size_t)QN * 64;
    float* S_p    = ws;                 ws += (size_t)LN * 64;   // 1 MB
    float* r_gate = ws;                 ws += (size_t)QN * 64;
    float* p_gate = ws;                 ws += (size_t)LN * 64;   // 1 MB
    float* prot   = ws;                 ws += 64;
    float* ctx    = ws;                 ws += (size_t)QN * 128;
    float* h1     = ws;                 ws += (size_t)QN * 256;
    float* h2     = ws;                 ws += (size_t)QN * 128;
    (void)ws_size; (void)in_sizes; (void)n_in; (void)out_size;

    // 1) p = relu(protein @ Wc^T + bc)          [4096,256]  (dominant GEMM)
    gemm(protein, Wc, bc, p, LN, 256, 1024, ACT_RELU, stream);
    // 2) r = relu(reactions @ W2^T + b2)        [512,64]
    gemm(reactions, W2, b2, r, QN, 64, 256, ACT_RELU, stream);
    // 3) k = relu(p @ W1^T + b1)                [4096,64]
    gemm(p, W1, b1, kbuf, LN, 64, 256, ACT_RELU, stream);
    // 4) ra = r @ Wra^T + bra                   [512,64]
    gemm(r, Wra, bra, ra, QN, 64, 64, ACT_NONE, stream);
    // 5) pa = k @ Wpa^T + bpa                   [4096,64]
    gemm(kbuf, Wpa, bpa, pa, LN, 64, 64, ACT_NONE, stream);

    // 6/7) collapsed [Q,L,64] tensor: means of relu outer-sum (Wa applied after,
    //      since the mean commutes with the linear projection @Wa^T + ba)
    relu_outer_mean<<<QN / 8, 256, 0, stream>>>(pa, LN, ra, S_r);   // [512,64]
    relu_outer_mean<<<LN / 8, 256, 0, stream>>>(ra, QN, pa, S_p);   // [4096,64]

    // 8/9) gates = sigmoid(S @ Wa^T + ba)
    gemm(S_r, Wa, ba, r_gate, QN, 64, 64, ACT_SIGMOID, stream);
    gemm(S_p, Wa, ba, p_gate, LN, 64, 64, ACT_SIGMOID, stream);

    // 10) prot = max_l k*(1+p_gate)             [64]
    prot_max_kernel<<<1, 256, 0, stream>>>(kbuf, p_gate, prot, LN);
    // 11) ctx = [ r*(1+r_gate) | prot ]         [512,128]
    ctx_kernel<<<(QN * 128) / 256, 256, 0, stream>>>(r, r_gate, prot, ctx, QN);

    // 12/13) MLP head
    gemm(ctx, Wf1, bf1, h1, QN, 256, 128, ACT_LRELU, stream);
    gemm(h1, Wf2, bf2, h2, QN, 128, 256, ACT_LRELU, stream);
    // 14) out = h2 @ Wf3^T + bf3                [512]
    out_kernel<<<(QN + 255) / 256, 256, 0, stream>>>(h2, Wf3, bf3, (float*)d_out, QN);
}